// roi_pool_22832046146312
// MI455X (gfx1250) — compile-verified
//
#include <hip/hip_runtime.h>
#include <cstdint>
#include <cfloat>

// ---------------------------------------------------------------------------
// RoI max pooling, MI455X (gfx1250).
//
// Roofline: features (5.12 MB) are L2-resident; output (12.8 MB) streams to
// HBM once; ~16 fmax per output element. Pure memory-bound => no WMMA
// structure exists for this op. CDNA5-specific choices:
//   * wave32 sizing: 256-thread blocks = 8 waves.
//   * per-block-uniform ROI index -> ROI descriptor loads scalarize (s_load).
//   * global_prefetch_b8 on the next feature row.
//   * non-temporal stores for the streamed output (preserve L2 for features).
//   * a compile-probe kernel (never launched) exercising the async-to-LDS
//     path (global_load_async_to_lds_b32 + s_wait_asynccnt) to validate the
//     CDNA5 ASYNCcnt asm syntax for a future LDS-staged variant.
// ---------------------------------------------------------------------------

namespace {
constexpr int B_  = 2;
constexpr int C_  = 256;
constexpr int H_  = 50;
constexpr int W_  = 50;
constexpr int R_  = 256;
constexpr int PH_ = 7;
constexpr int PW_ = 7;
constexpr int KH_ = (H_ + PH_ - 1) / PH_ + 1;   // 9  (matches -(-H//PH)+1)
constexpr int KW_ = (W_ + PW_ - 1) / PW_ + 1;   // 9
constexpr int BINS   = PH_ * PW_;               // 49
constexpr int BLOCK  = 256;                     // 8 wave32 waves
constexpr int OUT_N  = R_ * C_ * BINS;          // 3,211,264
constexpr int BLOCKS_PER_ROI = (C_ * BINS) / BLOCK;  // 49 (exact)
constexpr int NBLOCKS = OUT_N / BLOCK;          // 12544
static_assert((C_ * BINS) % BLOCK == 0, "block must tile one ROI exactly");
}

__global__ __launch_bounds__(BLOCK) void roi_maxpool_f32(
    const float* __restrict__ features,
    const float* __restrict__ rois,
    const float* __restrict__ scale_ptr,
    float* __restrict__ out)
{
    // One ROI per 49 consecutive blocks -> r is uniform per block (SGPR).
    const int r     = blockIdx.x / BLOCKS_PER_ROI;
    const int local = (blockIdx.x % BLOCKS_PER_ROI) * BLOCK + (int)threadIdx.x;
    const int c     = local / BINS;
    const int bin   = local - c * BINS;
    const int ph    = bin / PW_;
    const int pw    = bin - ph * PW_;

    const float scale = scale_ptr[0];
    const float* roi  = rois + (size_t)r * 5;   // uniform -> scalar loads
    const int b    = (int)roi[0];               // trunc, matches astype(int32)
    const int rs_w = (int)rintf(roi[1] * scale);  // rintf == round-half-even
    const int rs_h = (int)rintf(roi[2] * scale);
    const int re_w = (int)rintf(roi[3] * scale);
    const int re_h = (int)rintf(roi[4] * scale);

    const float roi_h = (float)max(re_h - rs_h + 1, 1);
    const float roi_w = (float)max(re_w - rs_w + 1, 1);
    const float bh = roi_h / (float)PH_;        // true divide, matches jnp
    const float bw = roi_w / (float)PW_;

    int hstart = (int)floorf((float)ph * bh) + rs_h;
    int hend   = (int)ceilf(((float)ph + 1.0f) * bh) + rs_h;
    hstart = min(max(hstart, 0), H_);
    hend   = min(max(hend,   0), H_);

    int wstart = (int)floorf((float)pw * bw) + rs_w;
    int wend   = (int)ceilf(((float)pw + 1.0f) * bw) + rs_w;
    wstart = min(max(wstart, 0), W_);
    wend   = min(max(wend,   0), W_);

    const bool empty = (hend <= hstart) || (wend <= wstart);
    // Reference only scans KH/KW candidates past start (always covers window
    // in this config, but cap anyway for bit-exact semantics).
    const int hstop = min(hend, hstart + KH_);
    const int wstop = min(wend, wstart + KW_);

    float m = -FLT_MAX;   // == jnp.finfo(float32).min
    if (!empty) {
        const float* plane = features + ((size_t)b * C_ + c) * (size_t)(H_ * W_);
        for (int h = hstart; h < hstop; ++h) {
            const float* row = plane + h * W_;
            // Pull next row toward the WGP while we reduce this one
            // (emits global_prefetch_b8; speculative, OOB-safe).
            __builtin_prefetch(row + W_, 0, 0);
            for (int w = wstart; w < wstop; ++w)
                m = fmaxf(m, row[w]);
        }
    }

    // Output flat index ((r*C + c)*PH + ph)*PW + pw == global thread index.
    const int oidx = blockIdx.x * BLOCK + (int)threadIdx.x;
    __builtin_nontemporal_store(empty ? 0.0f : m, out + oidx);
}

// ---------------------------------------------------------------------------
// Compile probe (NOT launched): validate CDNA5 async-to-LDS asm syntax so a
// later round can stage ROI windows into LDS through the ASYNCcnt path.
// ---------------------------------------------------------------------------
__global__ __launch_bounds__(32) void cdna5_async_probe(
    const float* __restrict__ g, float* __restrict__ o)
{
    __shared__ float buf[32];
    const uint32_t lds_addr = (uint32_t)(uintptr_t)(&buf[threadIdx.x]);
    const float*   gaddr    = g + threadIdx.x;
    asm volatile("global_load_async_to_lds_b32 %0, %1, off"
                 :: "v"(lds_addr), "v"(gaddr) : "memory");
    asm volatile("s_wait_asynccnt 0" ::: "memory");
    __syncthreads();
    o[threadIdx.x] = buf[threadIdx.x];
}

extern "C" void kernel_launch(void* const* d_in, const int* in_sizes, int n_in,
                              void* d_out, int out_size, void* d_ws, size_t ws_size,
                              hipStream_t stream)
{
    const float* features = (const float*)d_in[0];  // (2,256,50,50) f32
    const float* rois     = (const float*)d_in[1];  // (256,5) f32
    const float* scale    = (const float*)d_in[2];  // scalar f32
    float*       out      = (float*)d_out;          // (256,256,7,7) f32

    roi_maxpool_f32<<<dim3(NBLOCKS), dim3(BLOCK), 0, stream>>>(
        features, rois, scale, out);
}